// ProductLayer_6047313953254
// MI455X (gfx1250) — compile-verified
//
#include <hip/hip_runtime.h>
#include <hip/hip_bf16.h>
#include <stdint.h>

// ---------------------------------------------------------------------------
// Sorted segment-sum:  out = segment_sum(x[ptrs], csr, N_OUT)
//   x:    4,194,304 f32 (16 MiB, kept L2-resident)
//   ptrs: 33,554,432 i32 (streamed once -> non-temporal)
//   csr:  33,554,432 i32, sorted (streamed once -> non-temporal)
//   out:  4,194,304 f32 (L2-resident accumulation target)
// Memory-bound (~290 MiB @ 23.3 TB/s ~= 13 us). Data path: CDNA5 async
// global->LDS copies (ASYNCcnt) double-buffered with NT temporal hint so the
// 256 MiB index stream does not evict x/out from the 192 MB L2;
// ds_load_b128 readback from a bank-conflict-free padded layout;
// global_atomic_add_f32 only for segments shared across thread boundaries,
// plain stores for exclusively-owned interior segments.
// ---------------------------------------------------------------------------

namespace cfg {
constexpr int TPB   = 256;   // threads per block (8 wave32)
constexpr int EPT   = 16;    // edges per thread per tile
constexpr int TILES = 2;     // double-buffered tiles per block
constexpr int STRIDE = 20;   // ints per thread LDS region (16 data + 4 pad)
                             // 80B stride -> conflict-free ds_load_b128
constexpr int ARR    = TPB * STRIDE;    // ints per array (csr or ptrs) per tile
constexpr int TILE_I = 2 * ARR;         // ints per tile buffer (csr + ptrs)
constexpr int TILE_EDGES = TPB * EPT;   // 4096 edges per tile
}

__device__ __forceinline__ unsigned lds_addr(const void* p) {
  // Generic (flat) pointers into LDS carry the LDS byte offset in addr[31:0].
  return (unsigned)(uintptr_t)p;
}

__device__ __forceinline__ void async_copy_b128_nt(unsigned lds_off,
                                                   const void* sbase,
                                                   int gbyte_off) {
  // GVS form: mem_addr = SGPR64 base + VGPR32 offset; dest = LDS byte addr.
  // TH_LOAD_NT: stream-once data must not displace x/out in L2.
  asm volatile("global_load_async_to_lds_b128 %0, %1, %2 th:TH_LOAD_NT"
               :
               : "v"(lds_off), "v"(gbyte_off), "s"(sbase)
               : "memory");
}

__device__ __forceinline__ void stage_tile(const int* __restrict__ csr,
                                           const int* __restrict__ ptrs,
                                           int* smem_tile, int t,
                                           long long edge0) {
  const int goff = (int)(edge0 * 4) + t * (cfg::EPT * 4);  // bytes, 16B aligned
  const unsigned lc = lds_addr(&smem_tile[t * cfg::STRIDE]);
  const unsigned lp = lds_addr(&smem_tile[cfg::ARR + t * cfg::STRIDE]);
#pragma unroll
  for (int i = 0; i < 4; ++i) {
    async_copy_b128_nt(lc + 16 * i, csr,  goff + 16 * i);
    async_copy_b128_nt(lp + 16 * i, ptrs, goff + 16 * i);
  }
}

__device__ __forceinline__ void process_tile(const float* __restrict__ x,
                                             float* __restrict__ out,
                                             const int* smem_tile, int t) {
  int cs[cfg::EPT], ps[cfg::EPT];
  const int* c = &smem_tile[t * cfg::STRIDE];
  const int* p = &smem_tile[cfg::ARR + t * cfg::STRIDE];
#pragma unroll
  for (int i = 0; i < 4; ++i) {  // ds_load_b128 x8, conflict-free layout
    *reinterpret_cast<int4*>(&cs[4 * i]) =
        *reinterpret_cast<const int4*>(&c[4 * i]);
    *reinterpret_cast<int4*>(&ps[4 * i]) =
        *reinterpret_cast<const int4*>(&p[4 * i]);
  }

  // Issue all gathers up front; x is L2-resident so these overlap well.
  float xs[cfg::EPT];
#pragma unroll
  for (int i = 0; i < cfg::EPT; ++i) xs[i] = x[ps[i]];

  // Run-length scan of the sorted segment ids.
  //  - first flush (head segment, may be shared with prev thread): atomic
  //  - intermediate flushes (segment fully inside this run): plain store
  //  - final flush (tail segment, may be shared with next thread): atomic
  float acc = xs[0];
  int   cur = cs[0];
  bool  interior = false;
#pragma unroll
  for (int i = 1; i < cfg::EPT; ++i) {
    if (cs[i] != cur) {
      if (interior) out[cur] = acc;
      else          unsafeAtomicAdd(&out[cur], acc);  // global_atomic_add_f32
      interior = true;
      cur = cs[i];
      acc = 0.0f;
    }
    acc += xs[i];
  }
  unsafeAtomicAdd(&out[cur], acc);
}

__global__ void __launch_bounds__(cfg::TPB)
segsum_kernel(const float* __restrict__ x, const int* __restrict__ ptrs,
              const int* __restrict__ csr, float* __restrict__ out) {
  __shared__ int smem[cfg::TILES * cfg::TILE_I];  // 80 KiB of the 320 KiB WGP LDS

  const int t = threadIdx.x;
  const long long base =
      (long long)blockIdx.x * (cfg::TILES * cfg::TILE_EDGES);

  // Fill the double buffer: 8 async ops per tile per wave (in-order completion).
  stage_tile(csr, ptrs, &smem[0],           t, base);
  stage_tile(csr, ptrs, &smem[cfg::TILE_I], t, base + cfg::TILE_EDGES);

  // Tile 0 ready once only tile 1's 8 async ops remain outstanding.
  asm volatile("s_wait_asynccnt 0x8" ::: "memory");
  process_tile(x, out, &smem[0], t);

  asm volatile("s_wait_asynccnt 0x0" ::: "memory");
  process_tile(x, out, &smem[cfg::TILE_I], t);
}

__global__ void __launch_bounds__(256)
zero_out_kernel(float4* __restrict__ out, int n4) {
  int i = blockIdx.x * 256 + threadIdx.x;
  if (i < n4) out[i] = make_float4(0.f, 0.f, 0.f, 0.f);
}

extern "C" void kernel_launch(void* const* d_in, const int* in_sizes, int n_in,
                              void* d_out, int out_size, void* d_ws,
                              size_t ws_size, hipStream_t stream) {
  const float* x    = (const float*)d_in[0];
  const int*   ptrs = (const int*)d_in[1];
  const int*   csr  = (const int*)d_in[2];
  float*       out  = (float*)d_out;

  const int E = in_sizes[1];  // 33,554,432 edges

  // 1) Clear poisoned output (segment sum accumulates into it).
  const int n4 = out_size / 4;  // out_size is a multiple of 4
  zero_out_kernel<<<(n4 + 255) / 256, 256, 0, stream>>>((float4*)out, n4);

  // 2) Gather + sorted segment-sum. E is an exact multiple of the block tile.
  const int edges_per_block = cfg::TILES * cfg::TILE_EDGES;  // 8192
  const int blocks = E / edges_per_block;                    // 4096
  segsum_kernel<<<blocks, cfg::TPB, 0, stream>>>(x, ptrs, csr, out);
}